// ChannelSelfAttention_7730941132915
// MI455X (gfx1250) — compile-verified
//
#include <hip/hip_runtime.h>

typedef __attribute__((ext_vector_type(2))) float v2f;
typedef __attribute__((ext_vector_type(4))) float v4f;
typedef __attribute__((ext_vector_type(8))) float v8f;

#define BDIM 4
#define CDIM 248
#define TDIM 500
#define DDIM 64
#define CP   256   // C padded to 16*16 tiles
#define NQ   62    // 248 floats = 62 float4 per wts row

// One block per (b,t). 256 threads = 8 waves (wave32).
__global__ __launch_bounds__(256) void csa_kernel(
    const float* __restrict__ x,
    const float* __restrict__ wq, const float* __restrict__ bq,
    const float* __restrict__ wk, const float* __restrict__ bk,
    const float* __restrict__ wv, const float* __restrict__ bv,
    float* __restrict__ out, float* __restrict__ wts)
{
    __shared__ float xv[CP];      // x[b, :, t], zero padded
    __shared__ float alpha[CP];   // per-row softmax slope
    __shared__ float mrow[CP];    // per-row max of scores (closed form)
    __shared__ float sums[CP];    // Σ_e E   (later: 1/Σ)   (also max-scratch)
    __shared__ float wsums[CP];   // Σ_e E·x (also min-scratch)
    __shared__ float mbar[CP];    // per-row weighted mean, then reduced
    __shared__ float wqs[DDIM], wks[DDIM], bqs[DDIM], wvs[DDIM], bvs[DDIM];

    const int tid = threadIdx.x;
    const int t   = blockIdx.x;
    const int b   = blockIdx.y;
    const int bt  = b * TDIM + t;

    if (tid < DDIM) {
        wqs[tid] = wq[tid]; wks[tid] = wk[tid]; bqs[tid] = bq[tid];
        wvs[tid] = wv[tid]; bvs[tid] = bv[tid];
    }
    xv[tid] = (tid < CDIM) ? x[(b * CDIM + tid) * TDIM + t] : 0.0f;
    __syncthreads();

    // A = wq.wk, Cs = bq.wk  (redundant per-thread; LDS broadcast reads)
    float A = 0.f, Cs = 0.f;
    for (int d = 0; d < DDIM; ++d) {
        A  = fmaf(wqs[d], wks[d], A);
        Cs = fmaf(bqs[d], wks[d], Cs);
    }

    // xmax / xmin tree reduction (sums/wsums as scratch)
    sums[tid]  = (tid < CDIM) ? xv[tid] : -3.4e38f;
    wsums[tid] = (tid < CDIM) ? xv[tid] :  3.4e38f;
    __syncthreads();
    for (int s = 128; s > 0; s >>= 1) {
        if (tid < s) {
            sums[tid]  = fmaxf(sums[tid],  sums[tid + s]);
            wsums[tid] = fminf(wsums[tid], wsums[tid + s]);
        }
        __syncthreads();
    }
    const float xmax = sums[0];
    const float xmin = wsums[0];
    __syncthreads();

    // slope and exact row-max: scores(c,e) = alpha_c * x_e (+const dropped)
    {
        const float xc = xv[tid];
        const float a  = (tid < CDIM) ? (fmaf(A, xc, Cs) * 0.125f) : 0.0f;
        alpha[tid] = a;
        mrow[tid]  = (a >= 0.f) ? a * xmax : a * xmin;
    }
    __syncthreads();

    // ---- Pass A: WMMA f32 16x16x4 row reductions ----
    // A tile: E[c, e..e+3]; B: col0 = 1, col1 = x_e  ->  D col0 = ΣE, col1 = ΣE·x
    const int wave = tid >> 5;
    const int lane = tid & 31;
    const int n    = lane & 15;
    const int koff = (lane < 16) ? 0 : 2;   // 32-bit A-matrix 16x4 lane split
    for (int tile = 2 * wave; tile < 2 * wave + 2; ++tile) {
        const int   c0  = tile * 16;
        const float ac  = alpha[c0 + n];
        const float mc  = mrow[c0 + n];
        v8f acc = {};
        for (int k = 0; k < 62; ++k) {       // 62 * K4 = 248 columns, exact
            const int   e0  = 4 * k + koff;
            const float xe0 = xv[e0];
            const float xe1 = xv[e0 + 1];
            v2f am, bm;
            am.x = __expf(fmaf(ac, xe0, -mc));
            am.y = __expf(fmaf(ac, xe1, -mc));
            bm.x = (n == 0) ? 1.0f : ((n == 1) ? xe0 : 0.0f);
            bm.y = (n == 0) ? 1.0f : ((n == 1) ? xe1 : 0.0f);
            acc = __builtin_amdgcn_wmma_f32_16x16x4_f32(
                false, am, false, bm, (short)0, acc, false, false);
        }
        // C/D layout: lane<16 holds col N=lane rows 0..7; lane>=16 rows 8..15
        const int rbase = c0 + ((lane < 16) ? 0 : 8);
        if (n == 0) {
            #pragma unroll
            for (int j = 0; j < 8; ++j) sums[rbase + j] = acc[j];
        } else if (n == 1) {
            #pragma unroll
            for (int j = 0; j < 8; ++j) wsums[rbase + j] = acc[j];
        }
    }
    __syncthreads();

    // reciprocal row sums + per-row weighted mean
    {
        const float inv = 1.0f / sums[tid];
        sums[tid] = inv;
        mbar[tid] = (tid < CDIM) ? wsums[tid] * inv : 0.0f;
    }
    __syncthreads();
    for (int s = 128; s > 0; s >>= 1) {
        if (tid < s) mbar[tid] += mbar[tid + s];
        __syncthreads();
    }
    const float mean_m = mbar[0] * (1.0f / (float)CDIM);

    // out[b,t,:] = mean_m * wv + bv
    if (tid < DDIM) out[bt * DDIM + tid] = fmaf(mean_m, wvs[tid], bvs[tid]);

    // ---- Pass B: stream wts with 128b non-temporal stores ----
    // Row base is 16B aligned (248*4 = 62*16). wts is write-once (2.5x L2) -> NT.
    for (int c = wave; c < CDIM; c += 8) {
        const float ac  = alpha[c];
        const float mc  = mrow[c];
        const float inv = sums[c];
        v4f* __restrict__ rowq =
            reinterpret_cast<v4f*>(wts + (size_t)bt * (CDIM * CDIM) + c * CDIM);
        for (int q = lane; q < NQ; q += 32) {
            const int e = q * 4;
            v4f vv;
            vv.x = __expf(fmaf(ac, xv[e + 0], -mc)) * inv;
            vv.y = __expf(fmaf(ac, xv[e + 1], -mc)) * inv;
            vv.z = __expf(fmaf(ac, xv[e + 2], -mc)) * inv;
            vv.w = __expf(fmaf(ac, xv[e + 3], -mc)) * inv;
            __builtin_nontemporal_store(vv, rowq + q);
        }
    }
}

extern "C" void kernel_launch(void* const* d_in, const int* in_sizes, int n_in,
                              void* d_out, int out_size, void* d_ws, size_t ws_size,
                              hipStream_t stream) {
    const float* x  = (const float*)d_in[0];
    const float* wq = (const float*)d_in[1];
    const float* bq = (const float*)d_in[2];
    const float* wk = (const float*)d_in[3];
    const float* bk = (const float*)d_in[4];
    const float* wv = (const float*)d_in[5];
    const float* bv = (const float*)d_in[6];
    float* out = (float*)d_out;
    float* wts = out + (size_t)BDIM * TDIM * DDIM;   // tuple order: (out, wts)
    dim3 grid(TDIM, BDIM);
    csa_kernel<<<grid, 256, 0, stream>>>(x, wq, bq, wk, bk, wv, bv, out, wts);
}